// Model_lstm_37933151158694
// MI455X (gfx1250) — compile-verified
//
#include <hip/hip_runtime.h>

#ifndef __has_builtin
#define __has_builtin(x) 0
#endif

// ---- problem shape (fixed by the reference) ----
constexpr int kLag = 365;
constexpr int kXW  = 2 * kLag;   // 730 floats of x per batch row
constexpr int kS   = 256;        // states == threads per block

// ---- CDNA5 feature detection ----
#if __has_builtin(__builtin_amdgcn_global_load_async_to_lds_b32)
#define USE_ASYNC_LDS 1
#else
#define USE_ASYNC_LDS 0
#endif

#if __has_builtin(__builtin_amdgcn_tanhf)
#define HW_TANH 1
#else
#define HW_TANH 0
#endif

typedef __attribute__((address_space(1))) int gint_t;
typedef __attribute__((address_space(3))) int lint_t;

// sigma(z) = 0.5*tanh(z/2)+0.5  (1 trans op with v_tanh_f32)
__device__ __forceinline__ float fast_sigmoid(float z) {
#if HW_TANH
  return fmaf(0.5f, __builtin_amdgcn_tanhf(0.5f * z), 0.5f);
#else
  // 1/(1+exp(-z)) via exp2 + rcp
  float e = __builtin_amdgcn_exp2f(z * -1.442695040888963f);
  return __builtin_amdgcn_rcpf(1.0f + e);
#endif
}

__device__ __forceinline__ float fast_tanh(float z) {
#if HW_TANH
  return __builtin_amdgcn_tanhf(z);
#else
  // tanh(z) = (1-e)/(1+e), e = exp(-2z)
  float e = __builtin_amdgcn_exp2f(z * -2.885390081777927f);
  return (1.0f - e) * __builtin_amdgcn_rcpf(1.0f + e);
#endif
}

// One block per batch row b; thread j owns state j for all 365 steps.
__global__ __launch_bounds__(kS) void diag_lstm_kernel(
    const float* __restrict__ x,     // (B, 730)
    const float* __restrict__ wi,    // (S, 4, 2)
    const float* __restrict__ wr,    // (4, S)
    const float* __restrict__ bias,  // (4, S)
    const float* __restrict__ rw,    // (1, S)
    const float* __restrict__ rb,    // (1,)
    float* __restrict__ out, int B)
{
  __shared__ float sx[kXW];
  __shared__ float sred[kS];

  const int b = blockIdx.x;
  const int j = threadIdx.x;
  const float* gx = x + (size_t)b * kXW;

  // ---- stage this batch's x row into LDS (async-tensor path on CDNA5) ----
#if USE_ASYNC_LDS
  for (int i = j; i < kXW; i += kS) {
    __builtin_amdgcn_global_load_async_to_lds_b32(
        (gint_t*)(gx + i),
        (lint_t*)(sx + i),
        /*offset=*/0, /*cpol=*/0);
  }
#if __has_builtin(__builtin_amdgcn_s_wait_asynccnt)
  __builtin_amdgcn_s_wait_asynccnt(0);
#else
  asm volatile("s_wait_asynccnt 0" ::: "memory");
#endif
#else
  for (int i = j; i < kXW; i += kS) sx[i] = gx[i];
#endif
  __syncthreads();

  // ---- per-state constants (16 floats, loaded once) ----
  // weight_input[j,g,k]: 8 contiguous floats at j*8 (32B aligned)
  const float4* wiv = (const float4*)(wi + (size_t)j * 8);
  const float4 wA = wiv[0];  // {f:k0, f:k1, i:k0, i:k1}
  const float4 wB = wiv[1];  // {o:k0, o:k1, g:k0, g:k1}
  const float rf = wr[0 * kS + j], ri = wr[1 * kS + j];
  const float ro = wr[2 * kS + j], rg = wr[3 * kS + j];
  const float bf = bias[0 * kS + j], bi = bias[1 * kS + j];
  const float bo = bias[2 * kS + j], bg = bias[3 * kS + j];

  // ---- serial recurrence: 365 steps, 4 independent gate chains for ILP ----
  float h = 0.0f, c = 0.0f;
#pragma unroll 5
  for (int t = 0; t < kLag; ++t) {
    // step t uses x columns (lag-1-t, 2*lag-1-t); broadcast LDS reads
    const float x0 = sx[kLag - 1 - t];
    const float x1 = sx[2 * kLag - 1 - t];
    const float zf = fmaf(x1, wA.y, fmaf(x0, wA.x, fmaf(h, rf, bf)));
    const float zi = fmaf(x1, wA.w, fmaf(x0, wA.z, fmaf(h, ri, bi)));
    const float zo = fmaf(x1, wB.y, fmaf(x0, wB.x, fmaf(h, ro, bo)));
    const float zg = fmaf(x1, wB.w, fmaf(x0, wB.z, fmaf(h, rg, bg)));
    const float f  = fast_sigmoid(zf);
    const float i_ = fast_sigmoid(zi);
    const float o  = fast_sigmoid(zo);
    const float g  = fast_tanh(zg);
    c = fmaf(f, c, i_ * g);
    h = o * fast_tanh(c);
  }

  // ---- outputs: q (B), h_n ((B+1)*S), c_n ((B+1)*S), concatenated ----
  float* out_q = out;
  float* out_h = out + B;
  float* out_c = out_h + (size_t)(B + 1) * kS;

  out_h[(size_t)(b + 1) * kS + j] = h;
  out_c[(size_t)(b + 1) * kS + j] = c;
  if (b == 0) {  // explicit zero initial-state row (d_out is poisoned)
    out_h[j] = 0.0f;
    out_c[j] = 0.0f;
  }

  // q[b] = dot(h_row, reg_w) + reg_b : block-wide LDS tree reduction
  sred[j] = h * rw[j];
  __syncthreads();
#pragma unroll
  for (int sft = kS / 2; sft > 0; sft >>= 1) {
    if (j < sft) sred[j] += sred[j + sft];
    __syncthreads();
  }
  if (j == 0) out_q[b] = sred[0] + rb[0];
}

extern "C" void kernel_launch(void* const* d_in, const int* in_sizes, int n_in,
                              void* d_out, int out_size, void* d_ws, size_t ws_size,
                              hipStream_t stream) {
  (void)n_in; (void)out_size; (void)d_ws; (void)ws_size;
  const float* x    = (const float*)d_in[0];
  const float* wi   = (const float*)d_in[1];
  const float* wr   = (const float*)d_in[2];
  const float* bias = (const float*)d_in[3];
  const float* rw   = (const float*)d_in[4];
  const float* rb   = (const float*)d_in[5];
  const int B = in_sizes[0] / kXW;  // 8192

  diag_lstm_kernel<<<B, kS, 0, stream>>>(x, wi, wr, bias, rw, rb,
                                         (float*)d_out, B);
}